// GATLayer_71743133712500
// MI455X (gfx1250) — compile-verified
//
#include <hip/hip_runtime.h>
#include <hip/hip_bf16.h>

// ---------------------------------------------------------------------------
// GAT layer (heads=1, self-loops) for gfx1250 / MI455X, wave32.
//   xp    = x @ W                         (WMMA f32 16x16x4, exact fp32)
//   a_src = xp @ att_src ; a_dst = xp @ att_dst
//   e     = leakyrelu(a_src[src] + a_dst[dst], 0.2)
//   segment softmax over dst, out = segsum(alpha * xp[src]) + bias
// ---------------------------------------------------------------------------

typedef __attribute__((ext_vector_type(2))) float v2f;
typedef __attribute__((ext_vector_type(8))) float v8f;

#define F_DIM 128
#define NEG_SLOPE 0.2f

// ---- monotone float <-> uint key (for atomic float max via atomicMax(u32)) ----
__device__ __forceinline__ unsigned fkey(float f) {
    unsigned u = __float_as_uint(f);
    return (u & 0x80000000u) ? ~u : (u | 0x80000000u);
}
__device__ __forceinline__ float funkey(unsigned k) {
    unsigned u = (k & 0x80000000u) ? (k & 0x7FFFFFFFu) : ~k;
    return __uint_as_float(u);
}

__device__ __forceinline__ float lrelu(float v) {
    return v > 0.0f ? v : NEG_SLOPE * v;
}

// edge id -> (src, dst) with implicit self-loops appended after the E real edges
__device__ __forceinline__ void edge_ends(const int* __restrict__ ei, int e, int E,
                                          int& src, int& dst) {
    if (e < E) { src = ei[e]; dst = ei[E + e]; }
    else       { src = e - E; dst = e - E;     }
}

// ---------------------------------------------------------------------------
// Kernel 1: xp = x @ W via V_WMMA_F32_16X16X4_F32. One wave per 16x16 tile.
// Fragment layouts (wave32):
//   A (16x4):  lanes 0-15 -> M=lane,   VGPR0=K0..  lanes 16-31 -> K=+2
//              lane: a.x = x[m0+l][k+2*half], a.y = +1  (contiguous float2)
//   B (4x16):  VGPR0: lanes0-15 row K=k+0..  -> b.x = W[k+half][n0+l]
//              VGPR1: -> b.y = W[k+2+half][n0+l]
//   C/D (16x16): VGPR r: lanes0-15 M=r, lanes16-31 M=r+8, N = l
// ---------------------------------------------------------------------------
__global__ void k_gemm_wmma(const float* __restrict__ x, const float* __restrict__ W,
                            float* __restrict__ xp, int M) {
    const int wid  = (blockIdx.x * blockDim.x + threadIdx.x) >> 5;
    const int lane = threadIdx.x & 31;
    const int ntN  = F_DIM / 16;                 // 8 tiles along N
    const int tm   = wid / ntN;
    const int tn   = wid % ntN;
    if (tm * 16 >= M) return;                    // uniform per wave

    const int m0 = tm * 16, n0 = tn * 16;
    const int half = lane >> 4, l = lane & 15;
    const int mr = min(m0 + l, M - 1);           // safe clamp (M%16==0 normally)
    const float* __restrict__ arow = x + (size_t)mr * F_DIM;

    v8f c = {};
    #pragma unroll
    for (int k = 0; k < F_DIM; k += 4) {
        v2f a, b;
        a.x = arow[k + 2 * half];
        a.y = arow[k + 2 * half + 1];
        b.x = W[(size_t)(k + half)     * F_DIM + n0 + l];
        b.y = W[(size_t)(k + 2 + half) * F_DIM + n0 + l];
        c = __builtin_amdgcn_wmma_f32_16x16x4_f32(
                /*neg_a=*/false, a, /*neg_b=*/false, b,
                /*c_mod=*/(short)0, c, /*reuse_a=*/false, /*reuse_b=*/false);
    }

    #pragma unroll
    for (int r = 0; r < 8; ++r) {
        int row = m0 + r + 8 * half;
        if (row < M) xp[(size_t)row * F_DIM + n0 + l] = c[r];
    }
}

// ---------------------------------------------------------------------------
// Kernel 2: per-node attention logits. One wave per node, lane owns 4 features.
// ---------------------------------------------------------------------------
__global__ void k_att_dots(const float* __restrict__ xp,
                           const float* __restrict__ att_src,
                           const float* __restrict__ att_dst,
                           float* __restrict__ a_src, float* __restrict__ a_dst,
                           int N) {
    const int n    = (blockIdx.x * blockDim.x + threadIdx.x) >> 5;
    const int lane = threadIdx.x & 31;
    if (n >= N) return;
    const float* __restrict__ row = xp + (size_t)n * F_DIM;
    float s = 0.0f, d = 0.0f;
    #pragma unroll
    for (int j = 0; j < F_DIM / 32; ++j) {
        int f = lane + 32 * j;
        float v = row[f];
        s = fmaf(v, att_src[f], s);
        d = fmaf(v, att_dst[f], d);
    }
    #pragma unroll
    for (int off = 16; off > 0; off >>= 1) {
        s += __shfl_xor(s, off, 32);
        d += __shfl_xor(d, off, 32);
    }
    if (lane == 0) { a_src[n] = s; a_dst[n] = d; }
}

// ---------------------------------------------------------------------------
// Kernel 3: init out = bias (broadcast), mkey = 0 (== -worst key), denom = 0.
// ---------------------------------------------------------------------------
__global__ void k_init(float* __restrict__ out, const float* __restrict__ bias,
                       unsigned* __restrict__ mkey, float* __restrict__ denom,
                       int N) {
    const long long tid = (long long)blockIdx.x * blockDim.x + threadIdx.x;
    const long long total = (long long)N * F_DIM;
    if (tid >= total) return;
    const int f = (int)(tid & (F_DIM - 1));
    out[tid] = bias[f];
    if (f == 0) {
        const int n = (int)(tid >> 7);
        mkey[n]  = 0u;       // below any real float key; self-loop guarantees a write
        denom[n] = 0.0f;
    }
}

// ---------------------------------------------------------------------------
// Kernel 4: segment max of e over dst (thread per edge, atomic u32-key max).
// ---------------------------------------------------------------------------
__global__ void k_edge_max(const int* __restrict__ ei,
                           const float* __restrict__ a_src,
                           const float* __restrict__ a_dst,
                           unsigned* __restrict__ mkey, int E, int EN) {
    const int e = blockIdx.x * blockDim.x + threadIdx.x;
    if (e >= EN) return;
    int src, dst; edge_ends(ei, e, E, src, dst);
    const float ev = lrelu(a_src[src] + a_dst[dst]);
    atomicMax(&mkey[dst], fkey(ev));
}

// ---------------------------------------------------------------------------
// Kernel 5: denom[dst] += exp(e - max) (thread per edge).
// ---------------------------------------------------------------------------
__global__ void k_edge_sum(const int* __restrict__ ei,
                           const float* __restrict__ a_src,
                           const float* __restrict__ a_dst,
                           const unsigned* __restrict__ mkey,
                           float* __restrict__ denom, int E, int EN) {
    const int e = blockIdx.x * blockDim.x + threadIdx.x;
    if (e >= EN) return;
    int src, dst; edge_ends(ei, e, E, src, dst);
    const float ev = lrelu(a_src[src] + a_dst[dst]);
    const float w  = __expf(ev - funkey(mkey[dst]));
    atomicAdd(&denom[dst], w);
}

// ---------------------------------------------------------------------------
// Kernel 6: weighted scatter. One wave per edge; lane owns 4 strided (stride 32)
// features -> coalesced 512B gather of xp[src] + global_atomic_add_f32 into
// out[dst]. xp (25.6MB) is L2-resident on MI455X (192MB L2), so both the
// random gather and the atomics stay on-chip.
// ---------------------------------------------------------------------------
__global__ void k_scatter(const int* __restrict__ ei,
                          const float* __restrict__ xp,
                          const float* __restrict__ a_src,
                          const float* __restrict__ a_dst,
                          const unsigned* __restrict__ mkey,
                          const float* __restrict__ denom,
                          float* __restrict__ out, int E, int EN) {
    const int e    = (blockIdx.x * blockDim.x + threadIdx.x) >> 5;
    const int lane = threadIdx.x & 31;
    if (e >= EN) return;
    int src, dst; edge_ends(ei, e, E, src, dst);

    const float ev    = lrelu(a_src[src] + a_dst[dst]);
    const float alpha = __expf(ev - funkey(mkey[dst])) / denom[dst];

    const float* __restrict__ srow = xp  + (size_t)src * F_DIM;
    float* __restrict__       orow = out + (size_t)dst * F_DIM;
    #pragma unroll
    for (int j = 0; j < F_DIM / 32; ++j) {
        const int f = lane + 32 * j;
        atomicAdd(&orow[f], alpha * srow[f]);
    }
}

// ---------------------------------------------------------------------------
extern "C" void kernel_launch(void* const* d_in, const int* in_sizes, int n_in,
                              void* d_out, int out_size, void* d_ws, size_t ws_size,
                              hipStream_t stream) {
    const float* x       = (const float*)d_in[0];
    const float* W       = (const float*)d_in[1];
    const float* att_src = (const float*)d_in[2];
    const float* att_dst = (const float*)d_in[3];
    const float* bias    = (const float*)d_in[4];
    const int*   ei      = (const int*)d_in[5];

    const int N  = in_sizes[0] / F_DIM;   // 50000
    const int E  = in_sizes[5] / 2;       // 800000
    const int EN = E + N;                 // edges incl. self loops

    float* out = (float*)d_out;

    // workspace layout (floats)
    float*    ws    = (float*)d_ws;
    float*    xp    = ws;                         size_t off = (size_t)N * F_DIM;
    float*    asrc  = ws + off;                   off += N;
    float*    adst  = ws + off;                   off += N;
    unsigned* mkey  = (unsigned*)(ws + off);      off += N;
    float*    denom = ws + off;

    // 1) GEMM: one wave per 16x16 tile, 8 waves per 256-thread block
    {
        const int tilesM = (N + 15) / 16;
        const int waves  = tilesM * (F_DIM / 16);
        const int blocks = (waves + 7) / 8;
        k_gemm_wmma<<<blocks, 256, 0, stream>>>(x, W, xp, N);
    }
    // 2) attention logits: one wave per node
    {
        const int blocks = (N + 7) / 8;
        k_att_dots<<<blocks, 256, 0, stream>>>(xp, att_src, att_dst, asrc, adst, N);
    }
    // 3) init out/bias, max keys, denominators
    {
        const long long total = (long long)N * F_DIM;
        const int blocks = (int)((total + 255) / 256);
        k_init<<<blocks, 256, 0, stream>>>(out, bias, mkey, denom, N);
    }
    // 4) segment max
    {
        const int blocks = (EN + 255) / 256;
        k_edge_max<<<blocks, 256, 0, stream>>>(ei, asrc, adst, mkey, E, EN);
    }
    // 5) segment sum of exp
    {
        const int blocks = (EN + 255) / 256;
        k_edge_sum<<<blocks, 256, 0, stream>>>(ei, asrc, adst, mkey, denom, E, EN);
    }
    // 6) weighted scatter: one wave per edge
    {
        const int blocks = (EN + 7) / 8;
        k_scatter<<<blocks, 256, 0, stream>>>(ei, xp, asrc, adst, mkey, denom,
                                              out, E, EN);
    }
}